// CTREmbedding_60696477827087
// MI455X (gfx1250) — compile-verified
//
#include <hip/hip_runtime.h>

typedef float v2f __attribute__((ext_vector_type(2)));
typedef float v8f __attribute__((ext_vector_type(8)));

#define Bn 4
#define Ln 100
#define Mn 2048
#define En 50
#define INV_SU (1.0f / 1000.0f)
#define INV_TU (1.0f / 86400.0f)

// out[b,l,m,e] = base[b,l,e] + ds[b,l,m] * slope[b,l,e]
// Each wave computes one 16(m) x 16(e) tile via V_WMMA_F32_16X16X4_F32 as a
// rank-1 outer product (K=0 column of A = ds, K=0 row of B = slope, C = base).
__global__ __launch_bounds__(256) void ctr_embed_wmma(
    const int* __restrict__ traj_location,   // [B,L] 1-based ids
    const float* __restrict__ mat2,          // [NLOC,M]
    const float* __restrict__ vec,           // [B,L]
    const int* __restrict__ traj_length,     // [B]
    const float* __restrict__ emb_su,        // [2,E]
    const float* __restrict__ emb_sl,        // [2,E]
    const float* __restrict__ emb_tu,        // [2,E]
    const float* __restrict__ emb_tl,        // [2,E]
    float* __restrict__ out)                 // [B,L,M,E]
{
    const int tid    = threadIdx.x;
    const int wave   = tid >> 5;
    const int lane   = tid & 31;
    const int lane16 = lane & 15;
    const int half   = lane >> 4;

    const int e0 = (wave & 3) << 4;                      // e tile: 0,16,32,48
    const int m0 = (blockIdx.x << 5) + ((wave >> 2) << 4); // m tile base

    const int bl = blockIdx.y;                           // flattened (b,l)
    const int b  = bl / Ln;
    const int l  = bl - b * Ln;

    // Uniform per block -> EXEC stays all-1s through the WMMA.
    const bool valid = (l < traj_length[b]);
    const int  v     = valid ? 1 : 0;
    const int  loc   = traj_location[bl] - 1;
    const float dt   = vec[bl];

    // Per-lane embedding column (clamp reads for the partial last e-tile;
    // those lanes are masked at the store).
    const int e  = e0 + lane16;
    const int ec = (e < En) ? e : (En - 1);
    const float esl = emb_sl[v * En + ec];
    const float esu = emb_su[v * En + ec];
    const float etl = emb_tl[v * En + ec];
    const float etu = emb_tu[v * En + ec];

    const float slope = (esu - esl) * INV_SU;
    const float base  = esl + etl + (etu - etl) * (dt * INV_TU);

    // ds for this wave's 16 m-rows (both half-waves fetch the same 16 -> L0 hit).
    float dsv = 0.0f;
    if (valid) dsv = mat2[(size_t)loc * Mn + (m0 + lane16)];

    // A (16x4 f32): VGPR0 lanes 0-15 hold (M=lane, K=0) = ds[m]; rest zero.
    // B (4x16 f32): VGPR0 lanes 0-15 hold (K=0, N=lane) = slope[e]; rest zero.
    // Zeroing everything outside half 0 / component 0 makes the result
    // independent of how K=1..3 slots are distributed across halves.
    v2f a;  a.x  = (half == 0) ? dsv   : 0.0f;  a.y  = 0.0f;
    v2f bm; bm.x = (half == 0) ? slope : 0.0f;  bm.y = 0.0f;

    // C/D layout: VGPR r, half h -> (M = m0 + 8h + r, N = e0 + lane16).
    // base depends only on N, so broadcast it down all 8 accumulator VGPRs.
    v8f c;
    #pragma unroll
    for (int r = 0; r < 8; ++r) c[r] = base;

    v8f d = __builtin_amdgcn_wmma_f32_16x16x4_f32(
        /*neg_a=*/false, a, /*neg_b=*/false, bm,
        /*c_mod=*/(short)0, c, /*reuse_a=*/false, /*reuse_b=*/false);

    if (e < En) {
        float* orow = out + ((size_t)bl * Mn + (m0 + (half << 3))) * En + e;
        #pragma unroll
        for (int r = 0; r < 8; ++r)
            orow[(size_t)r * En] = d[r];   // lanes 0-15: 64B-contiguous per row
    }
}

extern "C" void kernel_launch(void* const* d_in, const int* in_sizes, int n_in,
                              void* d_out, int out_size, void* d_ws, size_t ws_size,
                              hipStream_t stream) {
    const int*   traj_location = (const int*)d_in[0];
    const float* mat2          = (const float*)d_in[1];
    const float* vec           = (const float*)d_in[2];
    const int*   traj_length   = (const int*)d_in[3];
    const float* emb_su        = (const float*)d_in[4];
    const float* emb_sl        = (const float*)d_in[5];
    const float* emb_tu        = (const float*)d_in[6];
    const float* emb_tl        = (const float*)d_in[7];
    float* out = (float*)d_out;

    dim3 grid(Mn / 32, Bn * Ln, 1);   // (64, 400): 8 waves/block, 1 WMMA tile/wave
    dim3 block(256, 1, 1);
    hipLaunchKernelGGL(ctr_embed_wmma, grid, block, 0, stream,
                       traj_location, mat2, vec, traj_length,
                       emb_su, emb_sl, emb_tu, emb_tl, out);
}